// Local_GNN_14336600834196
// MI455X (gfx1250) — compile-verified
//
#include <hip/hip_runtime.h>
#include <hip/hip_bf16.h>
#include <math.h>

#define N_NODES 100000
#define N_EDGES 1600000
#define K_KEEP  90000
#define P_SORT  131072
#define EPSV    1e-12f

typedef __attribute__((ext_vector_type(2))) float v2f;
typedef __attribute__((ext_vector_type(8))) float v8f;

// ---------------- generic elementwise ----------------
__global__ void fill_f32(float* p, float v, int count) {
    int i = blockIdx.x * blockDim.x + threadIdx.x;
    if (i < count) p[i] = v;
}
__global__ void fill_i32(int* p, int v, int count) {
    int i = blockIdx.x * blockDim.x + threadIdx.x;
    if (i < count) p[i] = v;
}
__global__ void relu_k(float* p, int count) {
    int i = blockIdx.x * blockDim.x + threadIdx.x;
    if (i < count) p[i] = fmaxf(p[i], 0.0f);
}

// ---------------- WMMA fp32 GEMM: C[M,N] = A[M,K] * B[K,N] ----------------
// Requirements (all call sites satisfy them): M % 16 == 0, K % 4 == 0,
// N % 4 == 0, K*Npad <= 7168 floats of LDS.
// B is staged zero-padded into LDS with the CDNA5 async-copy path
// (global_load_async_to_lds_b128 + s_wait_asynccnt), so the inner loop is
// branch-free: global_load_b64 (A frag) + 2x ds_load (B frag) + v_wmma.
// One wave computes one 16x16 C tile with V_WMMA_F32_16X16X4_F32.
__global__ void wmma_gemm_f32(const float* __restrict__ A, const float* __restrict__ Bm,
                              float* __restrict__ C, int M, int N, int K) {
    __shared__ float sB[112 * 64];                 // 28 KB max (K<=112, Npad<=64)
    const int Np = (N + 15) & ~15;                 // padded column count

    // ---- zero-fill the pad columns (plain LDS stores) ----
    if (N < Np) {
        for (int t = threadIdx.x; t < K * (Np - N); t += blockDim.x) {
            int r = t / (Np - N), c = N + (t - r * (Np - N));
            sB[r * Np + c] = 0.0f;
        }
    }
    // ---- async-copy valid B data: 16B chunks, rows are multiples of 16B ----
    {
        const int chunks = (K * N) >> 2;           // float4 chunks
        const int cpr    = N >> 2;                 // chunks per row
        for (int ch = threadIdx.x; ch < chunks; ch += blockDim.x) {
            int r = ch / cpr, cc = ch - r * cpr;
            unsigned lds_addr = (unsigned)(unsigned long long)&sB[r * Np + cc * 4];
            unsigned long long gaddr = (unsigned long long)(Bm + ch * 4);
            asm volatile("global_load_async_to_lds_b128 %0, %1, off"
                         :: "v"(lds_addr), "v"(gaddr) : "memory");
        }
        asm volatile("s_wait_asynccnt 0" ::: "memory");
    }
    __syncthreads();

    int gtid   = blockIdx.x * blockDim.x + threadIdx.x;
    int wave   = gtid >> 5;
    int lane   = threadIdx.x & 31;
    int ntiles = (N + 15) >> 4;
    int mt     = wave / ntiles;
    int nt     = wave - mt * ntiles;
    if (mt * 16 >= M) return;                      // wave-uniform: EXEC all-1s at WMMA

    const int r0   = mt << 4, c0 = nt << 4;
    const int half = lane >> 4;                    // 0: K pair (0,1); 1: K pair (2,3)
    const int l16  = lane & 15;
    const int ar   = r0 + l16;                     // always < M (M % 16 == 0)
    const int bcl  = c0 + l16;                     // LDS column, always < Np

    const float* ap = A + (long)ar * K + (half << 1);
    v8f acc = {0.f, 0.f, 0.f, 0.f, 0.f, 0.f, 0.f, 0.f};
    for (int kk = 0; kk < K; kk += 4) {
        int ka = kk + (half << 1);
        v2f a; a.x = ap[kk]; a.y = ap[kk + 1];
        if (kk + 4 < K) __builtin_prefetch(ap + kk + 4, 0, 1);   // global_prefetch_b8
        v2f b; b.x = sB[ka * Np + bcl]; b.y = sB[(ka + 1) * Np + bcl];
        acc = __builtin_amdgcn_wmma_f32_16x16x4_f32(false, a, false, b,
                                                    (short)0, acc, false, false);
    }
    const int bc = c0 + l16;
    if (bc < N) {
        const int crow = r0 + (half << 3);
#pragma unroll
        for (int v = 0; v < 8; ++v)
            C[(long)(crow + v) * N + bc] = acc[v];
    }
}

// ---------------- GCN degree / norm ----------------
__global__ void deg_accum(const int* __restrict__ col, const float* __restrict__ ew,
                          float* deg, int E) {
    int e = blockIdx.x * blockDim.x + threadIdx.x;
    if (e < E) atomicAdd(&deg[col[e]], ew[e]);
}
__global__ void dinv_k(float* deg, int n) {   // in-place deg -> rsqrt(deg)
    int i = blockIdx.x * blockDim.x + threadIdx.x;
    if (i < n) { float d = deg[i]; deg[i] = (d > 0.0f) ? rsqrtf(d) : 0.0f; }
}
__global__ void norm_k(const int* __restrict__ row, const int* __restrict__ col,
                       const float* __restrict__ ew, const float* __restrict__ dinv,
                       float* nrm, int E) {
    int e = blockIdx.x * blockDim.x + threadIdx.x;
    if (e < E) nrm[e] = dinv[row[e]] * ew[e] * dinv[col[e]];
}

// out[i,f] = bias[f] + hlin[i,f] * dinv[i]^2   (self-loop term)
__global__ void mp_init(const float* __restrict__ hlin, const float* __restrict__ dinv,
                        const float* __restrict__ bias, float* out, int n, int F) {
    int t = blockIdx.x * blockDim.x + threadIdx.x;
    if (t >= n * F) return;
    int i = t / F, f = t - i * F;
    float di = dinv[i];
    out[t] = bias[f] + hlin[t] * di * di;
}
// out[col[e],f] += hlin[row[e],f] * nrm[e]
__global__ void mp_scatter(const float* __restrict__ hlin, const int* __restrict__ r,
                           const int* __restrict__ c, const float* __restrict__ nrm,
                           float* out, int E, int F) {
    int t = blockIdx.x * blockDim.x + threadIdx.x;
    int e = t / F;
    if (e >= E) return;
    int f = t - e * F;
    float nv = nrm[e];
    if (nv != 0.0f) atomicAdd(&out[c[e] * F + f], hlin[r[e] * F + f] * nv);
}

// ---------------- score linear (N=1 too small for WMMA) ----------------
__global__ void score_dot(const float* __restrict__ h2, const float* __restrict__ Wsc,
                          float* sl, int n) {
    int i = blockIdx.x * blockDim.x + threadIdx.x;
    if (i >= n) return;
    float s = 0.0f;
#pragma unroll
    for (int f = 0; f < 20; ++f) s += h2[i * 20 + f] * Wsc[f];
    sl[i] = s;
}

// ---------------- top-k via bitonic sort (descending, stable) ----------------
__global__ void topk_init(const float* __restrict__ score, float* key, int* idx,
                          int n, int P) {
    int i = blockIdx.x * blockDim.x + threadIdx.x;
    if (i >= P) return;
    key[i] = (i < n) ? score[i] : -3.402823466e38f;
    idx[i] = (i < n) ? i : 0x7FFFFFFF;
}
__global__ void bitonic_step(float* key, int* idx, int j, int k2, int P) {
    int i = blockIdx.x * blockDim.x + threadIdx.x;
    if (i >= P) return;
    int ixj = i ^ j;
    if (ixj <= i) return;
    float ka = key[i], kb = key[ixj];
    int   ia = idx[i], ib = idx[ixj];
    bool a_before_b = (ka > kb) || (ka == kb && ia < ib);  // descending, idx tie-break
    bool want_desc  = ((i & k2) == 0);
    if (a_before_b != want_desc) {
        key[i] = kb; key[ixj] = ka; idx[i] = ib; idx[ixj] = ia;
    }
}
__global__ void perm_out_k(const int* __restrict__ idx, float* dst, int k) {
    int j = blockIdx.x * blockDim.x + threadIdx.x;
    if (j < k) dst[j] = (float)idx[j];
}
__global__ void xp_k(const float* __restrict__ h2, const float* __restrict__ key,
                     const int* __restrict__ idx, float* xp, int k) {
    int t = blockIdx.x * blockDim.x + threadIdx.x;
    if (t >= k * 20) return;
    int j = t / 20, f = t - j * 20;
    xp[t] = h2[idx[j] * 20 + f] * tanhf(key[j]);
}
__global__ void newid_set_k(const int* __restrict__ idx, int* newid, int k) {
    int j = blockIdx.x * blockDim.x + threadIdx.x;
    if (j < k) newid[idx[j]] = j;
}
// remap edges; ew2 stored into nrm2; accumulate deg2 (pre-initialized to 1.0)
__global__ void filter_k(const int* __restrict__ row, const int* __restrict__ col,
                         const float* __restrict__ ew, const int* __restrict__ newid,
                         int* r2, int* c2, float* nrm2, float* deg2, int E) {
    int e = blockIdx.x * blockDim.x + threadIdx.x;
    if (e >= E) return;
    int r = newid[row[e]], c = newid[col[e]];
    bool valid = (r >= 0) && (c >= 0);
    int rr = valid ? r : 0, cc = valid ? c : 0;
    float w = valid ? ew[e] : 0.0f;
    r2[e] = rr; c2[e] = cc; nrm2[e] = w;
    if (w != 0.0f) atomicAdd(&deg2[cc], w);
}
__global__ void norm2_k(const int* __restrict__ r2, const int* __restrict__ c2,
                        const float* __restrict__ dinv2, float* nrm2, int E) {
    int e = blockIdx.x * blockDim.x + threadIdx.x;
    if (e < E) nrm2[e] = dinv2[r2[e]] * nrm2[e] * dinv2[c2[e]];
}

// ---------------- MI estimator ----------------
__global__ void mi_accum(const float* __restrict__ embed, const float* __restrict__ h2,
                         const int* __restrict__ negp, const float* __restrict__ Wfc,
                         const float* __restrict__ bfc, float* acc, int n) {
    __shared__ float sj[256], sm[256];
    int i = blockIdx.x * blockDim.x + threadIdx.x;
    float vj = 0.0f, vm = 0.0f;
    if (i < n) {
        float a = bfc[0], b = bfc[0];
        int np = negp[i];
#pragma unroll
        for (int f = 0; f < 20; ++f) {
            float e = embed[i * 20 + f];
            a += e * Wfc[f];
            b += e * Wfc[f];
            a += h2[i * 20 + f]  * Wfc[20 + f];
            b += h2[np * 20 + f] * Wfc[20 + f];
        }
        a = a / fmaxf(fabsf(a), EPSV);   // F.normalize on [n,1]
        b = b / fmaxf(fabsf(b), EPSV);
        vj = a; vm = expf(b);
    }
    sj[threadIdx.x] = vj; sm[threadIdx.x] = vm;
    __syncthreads();
    for (int s = 128; s > 0; s >>= 1) {
        if ((int)threadIdx.x < s) {
            sj[threadIdx.x] += sj[threadIdx.x + s];
            sm[threadIdx.x] += sm[threadIdx.x + s];
        }
        __syncthreads();
    }
    if (threadIdx.x == 0) { atomicAdd(&acc[0], sj[0]); atomicAdd(&acc[1], sm[0]); }
}
__global__ void mi_final(const float* __restrict__ acc, float* dst, float inv_n) {
    if (blockIdx.x == 0 && threadIdx.x == 0)
        dst[0] = acc[0] * inv_n - logf(acc[1] * inv_n);
}

__global__ void cat_out_k(const float* __restrict__ xp, const float* __restrict__ h3,
                          float* out, int count) {
    int t = blockIdx.x * blockDim.x + threadIdx.x;
    if (t < count) out[t] = xp[t] + h3[t];
}

// ---------------- host driver ----------------
extern "C" void kernel_launch(void* const* d_in, const int* in_sizes, int n_in,
                              void* d_out, int out_size, void* d_ws, size_t ws_size,
                              hipStream_t stream) {
    (void)in_sizes; (void)n_in; (void)out_size; (void)ws_size;
    const float* x    = (const float*)d_in[0];
    const int*   eidx = (const int*)  d_in[1];
    const float* ew   = (const float*)d_in[2];
    const int*   negp = (const int*)  d_in[3];
    const float* W1   = (const float*)d_in[4];
    const float* b1   = (const float*)d_in[5];
    const float* W2   = (const float*)d_in[6];
    const float* b2   = (const float*)d_in[7];
    const float* Wsc  = (const float*)d_in[8];
    const float* bsc  = (const float*)d_in[9];
    const float* Wg   = (const float*)d_in[10];
    const float* bg   = (const float*)d_in[11];
    const float* Wfc  = (const float*)d_in[12];
    const float* bfc  = (const float*)d_in[13];
    const float* W3   = (const float*)d_in[14];
    const float* b3   = (const float*)d_in[15];
    const int* row = eidx;
    const int* col = eidx + N_EDGES;
    float* out = (float*)d_out;

    // workspace layout (256B aligned, with lifetime-based reuse)
    char* ws = (char*)d_ws;
    size_t off = 0;
    auto alloc = [&](size_t bytes) -> char* {
        char* p = ws + off;
        off += (bytes + 255) & ~(size_t)255;
        return p;
    };
    float* dinv  = (float*)alloc((size_t)N_NODES * 4);        // deg -> dinv in place
    float* nrm   = (float*)alloc((size_t)N_EDGES * 4);
    float* h2    = (float*)alloc((size_t)N_NODES * 20 * 4);
    float* embed = (float*)alloc((size_t)N_NODES * 20 * 4);
    float* score = (float*)alloc((size_t)N_NODES * 4);
    int*   r2    = (int*)  alloc((size_t)N_EDGES * 4);
    int*   c2    = (int*)  alloc((size_t)N_EDGES * 4);
    float* nrm2  = (float*)alloc((size_t)N_EDGES * 4);
    char*  bufA  = alloc((size_t)N_NODES * 64 * 4);           // 25.6 MB
    char*  bufB  = alloc((size_t)N_NODES * 64 * 4);           // 25.6 MB
    // bufA aliases (h1lin dead before h2lin written, etc.)
    float* h1lin  = (float*)bufA;
    float* h2lin  = (float*)bufA;
    float* emblin = (float*)(bufA + 8000000);
    float* sclin  = (float*)(bufA + 16000000);
    float* h3lin  = (float*)(bufA + 16400128);
    // bufB aliases (h1 dead after GEMM2)
    float* h1    = (float*)bufB;
    float* xp    = (float*)bufB;
    float* h3    = (float*)(bufB + 7200000);
    float* key   = (float*)(bufB + 14400000);
    int*   sidx  = (int*)  (bufB + 14400000 + (size_t)P_SORT * 4);
    int*   newid = (int*)  (bufB + 14400000 + (size_t)P_SORT * 8);
    float* deg2  = (float*)(bufB + 15848704);
    float* acc   = (float*)(bufB + 16208896);

    const int B = 256;
#define GRID(x) dim3((unsigned)(((long)(x) + 255) / 256))
    auto gemm = [&](const float* A, const float* Bmat, float* C, int M, int N, int K) {
        long waves = (long)((M + 15) / 16) * ((N + 15) / 16);
        wmma_gemm_f32<<<GRID(waves * 32), B, 0, stream>>>(A, Bmat, C, M, N, K);
    };

    // ---- shared degree / norm (same graph for conv1,2,embed,score) ----
    fill_f32<<<GRID(N_NODES), B, 0, stream>>>(dinv, 1.0f, N_NODES);  // self-loop w=1
    deg_accum<<<GRID(N_EDGES), B, 0, stream>>>(col, ew, dinv, N_EDGES);
    dinv_k<<<GRID(N_NODES), B, 0, stream>>>(dinv, N_NODES);
    norm_k<<<GRID(N_EDGES), B, 0, stream>>>(row, col, ew, dinv, nrm, N_EDGES);

    // ---- conv1: h1 = relu(mp(x@W1)) ----
    gemm(x, W1, h1lin, N_NODES, 64, 112);
    mp_init<<<GRID(N_NODES * 64), B, 0, stream>>>(h1lin, dinv, b1, h1, N_NODES, 64);
    mp_scatter<<<GRID((long)N_EDGES * 64), B, 0, stream>>>(h1lin, row, col, nrm, h1, N_EDGES, 64);
    relu_k<<<GRID(N_NODES * 64), B, 0, stream>>>(h1, N_NODES * 64);

    // ---- conv2: h2 = relu(mp(h1@W2)) ----
    gemm(h1, W2, h2lin, N_NODES, 20, 64);
    mp_init<<<GRID(N_NODES * 20), B, 0, stream>>>(h2lin, dinv, b2, h2, N_NODES, 20);
    mp_scatter<<<GRID((long)N_EDGES * 20), B, 0, stream>>>(h2lin, row, col, nrm, h2, N_EDGES, 20);
    relu_k<<<GRID(N_NODES * 20), B, 0, stream>>>(h2, N_NODES * 20);

    // ---- embed = mp(h2@Wg) (no relu) ----
    gemm(h2, Wg, emblin, N_NODES, 20, 20);
    mp_init<<<GRID(N_NODES * 20), B, 0, stream>>>(emblin, dinv, bg, embed, N_NODES, 20);
    mp_scatter<<<GRID((long)N_EDGES * 20), B, 0, stream>>>(emblin, row, col, nrm, embed, N_EDGES, 20);

    // ---- MI ----
    fill_f32<<<dim3(1), B, 0, stream>>>(acc, 0.0f, 2);
    mi_accum<<<GRID(N_NODES), B, 0, stream>>>(embed, h2, negp, Wfc, bfc, acc, N_NODES);
    mi_final<<<dim3(1), 64, 0, stream>>>(acc, out + 1890000, 1.0f / (float)N_NODES);

    // ---- score = mp(h2@Wsc) ----
    score_dot<<<GRID(N_NODES), B, 0, stream>>>(h2, Wsc, sclin, N_NODES);
    mp_init<<<GRID(N_NODES), B, 0, stream>>>(sclin, dinv, bsc, score, N_NODES, 1);
    mp_scatter<<<GRID(N_EDGES), B, 0, stream>>>(sclin, row, col, nrm, score, N_EDGES, 1);

    // ---- top-k (descending bitonic sort over padded P) ----
    topk_init<<<GRID(P_SORT), B, 0, stream>>>(score, key, sidx, N_NODES, P_SORT);
    for (int k2 = 2; k2 <= P_SORT; k2 <<= 1)
        for (int j = k2 >> 1; j > 0; j >>= 1)
            bitonic_step<<<GRID(P_SORT), B, 0, stream>>>(key, sidx, j, k2, P_SORT);

    perm_out_k<<<GRID(K_KEEP), B, 0, stream>>>(sidx, out + 1800000, K_KEEP);
    xp_k<<<GRID(K_KEEP * 20), B, 0, stream>>>(h2, key, sidx, xp, K_KEEP);

    // ---- filter_adj ----
    fill_i32<<<GRID(N_NODES), B, 0, stream>>>(newid, -1, N_NODES);
    newid_set_k<<<GRID(K_KEEP), B, 0, stream>>>(sidx, newid, K_KEEP);
    fill_f32<<<GRID(K_KEEP), B, 0, stream>>>(deg2, 1.0f, K_KEEP);
    filter_k<<<GRID(N_EDGES), B, 0, stream>>>(row, col, ew, newid, r2, c2, nrm2, deg2, N_EDGES);
    dinv_k<<<GRID(K_KEEP), B, 0, stream>>>(deg2, K_KEEP);
    norm2_k<<<GRID(N_EDGES), B, 0, stream>>>(r2, c2, deg2, nrm2, N_EDGES);

    // ---- conv3 on pooled graph: h3 = relu(mp2(xp@W3)) ----
    gemm(xp, W3, h3lin, K_KEEP, 20, 20);
    mp_init<<<GRID(K_KEEP * 20), B, 0, stream>>>(h3lin, deg2, b3, h3, K_KEEP, 20);
    mp_scatter<<<GRID((long)N_EDGES * 20), B, 0, stream>>>(h3lin, r2, c2, nrm2, h3, N_EDGES, 20);
    relu_k<<<GRID(K_KEEP * 20), B, 0, stream>>>(h3, K_KEEP * 20);

    // ---- graph_embedding = (xp + h3).reshape(1,-1) ----
    cat_out_k<<<GRID(K_KEEP * 20), B, 0, stream>>>(xp, h3, out, K_KEEP * 20);
#undef GRID
}